// OrientedDGCNN_9663676416051
// MI455X (gfx1250) — compile-verified
//
#include <hip/hip_runtime.h>

#define BDIM 16
#define NPT  2048
#define KNN  20
#define NKC  (NPT * KNN)   // 40960 columns per batch

typedef float v2f __attribute__((ext_vector_type(2)));
typedef float v8f __attribute__((ext_vector_type(8)));

// D = A(16x4) * B(4x16) + C, fp32 WMMA (CDNA5)
__device__ __forceinline__ v8f wmma4(v2f a, v2f b, v8f c) {
  return __builtin_amdgcn_wmma_f32_16x16x4_f32(false, a, false, b, (short)0, c,
                                               false, false);
}

__device__ __forceinline__ float lrelu(float v) { return v > 0.f ? v : 0.2f * v; }

__device__ __forceinline__ void topk_insert(float v, int m, float* best, int* bi) {
  if (v > best[KNN - 1]) {
    int p = KNN - 1;
    while (p > 0 && v > best[p - 1]) {
      best[p] = best[p - 1];
      bi[p] = bi[p - 1];
      --p;
    }
    best[p] = v;
    bi[p] = m;
  }
}

// ---------------------------------------------------------------- xg = (x - p_center) @ R_center
__global__ void xg_kernel(const float* __restrict__ x, const float* __restrict__ pc,
                          const float* __restrict__ Rc, float* __restrict__ xg) {
  int gid = blockIdx.x * 256 + threadIdx.x;  // b*NPT + n
  if (gid >= BDIM * NPT) return;
  int b = gid / NPT;
  float d0 = x[gid * 3 + 0] - pc[b * 3 + 0];
  float d1 = x[gid * 3 + 1] - pc[b * 3 + 1];
  float d2 = x[gid * 3 + 2] - pc[b * 3 + 2];
  const float* R = Rc + b * 9;
#pragma unroll
  for (int i = 0; i < 3; ++i)
    xg[gid * 3 + i] = d0 * R[0 * 3 + i] + d1 * R[1 * 3 + i] + d2 * R[2 * 3 + i];
}

// ---------------------------------------------------------------- kNN on 3-D coords
__global__ void knn3_kernel(const float* __restrict__ x, int* __restrict__ idx) {
  int b = blockIdx.y;
  int n = blockIdx.x * 256 + threadIdx.x;
  const float* xb = x + (size_t)b * NPT * 3;
  float cx = xb[n * 3 + 0], cy = xb[n * 3 + 1], cz = xb[n * 3 + 2];
  float cc = cx * cx + cy * cy + cz * cz;
  float best[KNN];
  int bi[KNN];
#pragma unroll
  for (int i = 0; i < KNN; ++i) { best[i] = -__builtin_inff(); bi[i] = 0; }
  __shared__ float sx[256], sy[256], sz[256], ss[256];
  for (int t = 0; t < NPT; t += 256) {
    __syncthreads();
    int m = t + threadIdx.x;
    float px = xb[m * 3 + 0], py = xb[m * 3 + 1], pz = xb[m * 3 + 2];
    sx[threadIdx.x] = px; sy[threadIdx.x] = py; sz[threadIdx.x] = pz;
    ss[threadIdx.x] = px * px + py * py + pz * pz;
    __syncthreads();
    for (int j = 0; j < 256; ++j) {
      float inner = cx * sx[j] + cy * sy[j] + cz * sz[j];
      float v = 2.f * inner - cc - ss[j];
      topk_insert(v, t + j, best, bi);
    }
  }
  int* op = idx + ((size_t)b * NPT + n) * KNN;
#pragma unroll
  for (int i = 0; i < KNN; ++i) op[i] = bi[i];
}

// ---------------------------------------------------------------- kNN on 64-D features (layout [b][c][n])
__global__ void knn64_kernel(const float* __restrict__ f, int* __restrict__ idx) {
  int b = blockIdx.y;
  int n = blockIdx.x * 128 + threadIdx.x;
  const float* fb = f + (size_t)b * 64 * NPT;
  float cen[64];
  float cc = 0.f;
#pragma unroll
  for (int c = 0; c < 64; ++c) {
    float v = fb[(size_t)c * NPT + n];
    cen[c] = v;
    cc += v * v;
  }
  __shared__ float sf[64][129];
  __shared__ float ss[128];
  float best[KNN];
  int bi[KNN];
#pragma unroll
  for (int i = 0; i < KNN; ++i) { best[i] = -__builtin_inff(); bi[i] = 0; }
  for (int t = 0; t < NPT; t += 128) {
    __syncthreads();
    {
      float s2 = 0.f;
#pragma unroll
      for (int c = 0; c < 64; ++c) {
        float v = fb[(size_t)c * NPT + t + threadIdx.x];
        sf[c][threadIdx.x] = v;
        s2 += v * v;
      }
      ss[threadIdx.x] = s2;
    }
    __syncthreads();
    for (int j = 0; j < 128; ++j) {
      float inner = 0.f;
#pragma unroll
      for (int c = 0; c < 64; ++c) inner += cen[c] * sf[c][j];
      float v = 2.f * inner - cc - ss[j];
      topk_insert(v, t + j, best, bi);
    }
  }
  int* op = idx + ((size_t)b * NPT + n) * KNN;
#pragma unroll
  for (int i = 0; i < KNN; ++i) op[i] = bi[i];
}

// ---------------------------------------------------------------- build 12-channel edge features
__global__ void build_h12_kernel(const float* __restrict__ x, const float* __restrict__ xg,
                                 const float* __restrict__ R, const int* __restrict__ idx,
                                 float* __restrict__ h12) {
  long gid = (long)blockIdx.x * 256 + threadIdx.x;  // b*NPT*KNN + n*KNN + k
  if (gid >= (long)BDIM * NPT * KNN) return;
  long t = gid / KNN;
  int n = (int)(t % NPT);
  int b = (int)(t / NPT);
  int m = idx[gid];
  const float* xb = x + (size_t)b * NPT * 3;
  float d0 = xb[m * 3 + 0] - xb[n * 3 + 0];
  float d1 = xb[m * 3 + 1] - xb[n * 3 + 1];
  float d2 = xb[m * 3 + 2] - xb[n * 3 + 2];
  const float* Rp = R + ((size_t)b * NPT + n) * 18;  // [2][3][3]
  size_t col = (size_t)n * KNN + (gid % KNN);
  float* hb = h12 + (size_t)b * 12 * NKC;
#pragma unroll
  for (int fd = 0; fd < 2; ++fd)
#pragma unroll
    for (int i = 0; i < 3; ++i) {
      float v = d0 * Rp[fd * 9 + 0 + i] + d1 * Rp[fd * 9 + 3 + i] + d2 * Rp[fd * 9 + 6 + i];
      hb[(size_t)(fd * 3 + i) * NKC + col] = v;
    }
  const float* gb = xg + (size_t)b * NPT * 3;
#pragma unroll
  for (int i = 0; i < 3; ++i) {
    hb[(size_t)(6 + i) * NKC + col] = gb[m * 3 + i] - gb[n * 3 + i];
    hb[(size_t)(9 + i) * NKC + col] = gb[n * 3 + i];
  }
}

// ---------------------------------------------------------------- stage1: h12 -> conv(w1) -> conv(w2) -> max_k
__global__ void stage1_kernel(const float* __restrict__ h12, const float* __restrict__ w1,
                              const float* __restrict__ s1, const float* __restrict__ b1,
                              const float* __restrict__ w2, const float* __restrict__ s2,
                              const float* __restrict__ b2, float* __restrict__ x1) {
  int b = blockIdx.y;
  int n0 = blockIdx.x * 16;
  int tid = threadIdx.x;         // 0..127
  int wave = tid >> 5;           // 0..3
  int lane = tid & 31;
  int l15 = lane & 15;
  int kh = (lane < 16) ? 0 : 2;
  int hoff = (lane < 16) ? 0 : 8;
  __shared__ float act1[64][17];
  __shared__ float act2[64][17];
  __shared__ float xm[64][16];
  for (int i = tid; i < 64 * 16; i += 128) ((float*)xm)[i] = -__builtin_inff();
  const float* hb = h12 + (size_t)b * 12 * NKC + (size_t)n0 * KNN;
  for (int jt = 0; jt < 20; ++jt) {
    int cl = jt * 16 + l15;
    v8f acc;
#pragma unroll
    for (int i = 0; i < 8; ++i) acc[i] = 0.f;
#pragma unroll
    for (int kk = 0; kk < 3; ++kk) {
      v2f a, bf;
      int mA = wave * 16 + l15;
      a[0] = w1[mA * 12 + kk * 4 + kh + 0];
      a[1] = w1[mA * 12 + kk * 4 + kh + 1];
      bf[0] = hb[(size_t)(kk * 4 + kh + 0) * NKC + cl];
      bf[1] = hb[(size_t)(kk * 4 + kh + 1) * NKC + cl];
      acc = wmma4(a, bf, acc);
    }
    __syncthreads();  // previous iteration's act1 readers done
#pragma unroll
    for (int r = 0; r < 8; ++r) {
      int row = wave * 16 + r + hoff;
      act1[row][l15] = lrelu(acc[r] * s1[row] + b1[row]);
    }
    __syncthreads();
    v8f acc2;
#pragma unroll
    for (int i = 0; i < 8; ++i) acc2[i] = 0.f;
#pragma unroll
    for (int kk = 0; kk < 16; ++kk) {
      v2f a, bf;
      int mA = wave * 16 + l15;
      a[0] = w2[mA * 64 + kk * 4 + kh + 0];
      a[1] = w2[mA * 64 + kk * 4 + kh + 1];
      bf[0] = act1[kk * 4 + kh + 0][l15];
      bf[1] = act1[kk * 4 + kh + 1][l15];
      acc2 = wmma4(a, bf, acc2);
    }
#pragma unroll
    for (int r = 0; r < 8; ++r) {
      int row = wave * 16 + r + hoff;
      act2[row][l15] = lrelu(acc2[r] * s2[row] + b2[row]);
    }
    __syncthreads();
    if (tid < 64) {
      int c0 = jt * 16;
      for (int j = 0; j < 16; ++j) {
        int p = (c0 + j) / KNN;
        xm[tid][p] = fmaxf(xm[tid][p], act2[tid][j]);
      }
    }
    __syncthreads();
  }
  for (int i = tid; i < 64 * 16; i += 128) {
    int row = i >> 4;
    int p = i & 15;
    x1[((size_t)b * 64 + row) * NPT + n0 + p] = xm[row][p];
  }
}

// ---------------------------------------------------------------- stage2: graph_feature(xin) -> conv(wA 64x128) -> conv(wB 64x64) -> max_k
__global__ void stage2_kernel(const float* __restrict__ xin, const int* __restrict__ idx,
                              const float* __restrict__ wA, const float* __restrict__ sA,
                              const float* __restrict__ bA, const float* __restrict__ wB,
                              const float* __restrict__ sB, const float* __restrict__ bB,
                              float* __restrict__ xout) {
  int b = blockIdx.y;
  int n0 = blockIdx.x * 16;
  int tid = threadIdx.x;
  int wave = tid >> 5;
  int lane = tid & 31;
  int l15 = lane & 15;
  int kh = (lane < 16) ? 0 : 2;
  int hoff = (lane < 16) ? 0 : 8;
  __shared__ float act1[64][17];
  __shared__ float act2[64][17];
  __shared__ float xm[64][16];
  for (int i = tid; i < 64 * 16; i += 128) ((float*)xm)[i] = -__builtin_inff();
  const float* xb = xin + (size_t)b * 64 * NPT;
  for (int jt = 0; jt < 20; ++jt) {
    int cl = jt * 16 + l15;
    int p = cl / KNN;
    int k = cl % KNN;
    int n = n0 + p;
    int m = idx[((size_t)b * NPT + n) * KNN + k];
    v8f acc;
#pragma unroll
    for (int i = 0; i < 8; ++i) acc[i] = 0.f;
#pragma unroll
    for (int kk = 0; kk < 16; ++kk) {  // channels 0..63: neighbor - center
      v2f a, bf;
      int mA = wave * 16 + l15;
      a[0] = wA[mA * 128 + kk * 4 + kh + 0];
      a[1] = wA[mA * 128 + kk * 4 + kh + 1];
      int c0 = kk * 4 + kh;
      bf[0] = xb[(size_t)(c0 + 0) * NPT + m] - xb[(size_t)(c0 + 0) * NPT + n];
      bf[1] = xb[(size_t)(c0 + 1) * NPT + m] - xb[(size_t)(c0 + 1) * NPT + n];
      acc = wmma4(a, bf, acc);
    }
#pragma unroll
    for (int kk = 16; kk < 32; ++kk) {  // channels 64..127: center
      v2f a, bf;
      int mA = wave * 16 + l15;
      a[0] = wA[mA * 128 + kk * 4 + kh + 0];
      a[1] = wA[mA * 128 + kk * 4 + kh + 1];
      int c0 = (kk - 16) * 4 + kh;
      bf[0] = xb[(size_t)(c0 + 0) * NPT + n];
      bf[1] = xb[(size_t)(c0 + 1) * NPT + n];
      acc = wmma4(a, bf, acc);
    }
    __syncthreads();
#pragma unroll
    for (int r = 0; r < 8; ++r) {
      int row = wave * 16 + r + hoff;
      act1[row][l15] = lrelu(acc[r] * sA[row] + bA[row]);
    }
    __syncthreads();
    v8f acc2;
#pragma unroll
    for (int i = 0; i < 8; ++i) acc2[i] = 0.f;
#pragma unroll
    for (int kk = 0; kk < 16; ++kk) {
      v2f a, bf;
      int mA = wave * 16 + l15;
      a[0] = wB[mA * 64 + kk * 4 + kh + 0];
      a[1] = wB[mA * 64 + kk * 4 + kh + 1];
      bf[0] = act1[kk * 4 + kh + 0][l15];
      bf[1] = act1[kk * 4 + kh + 1][l15];
      acc2 = wmma4(a, bf, acc2);
    }
#pragma unroll
    for (int r = 0; r < 8; ++r) {
      int row = wave * 16 + r + hoff;
      act2[row][l15] = lrelu(acc2[r] * sB[row] + bB[row]);
    }
    __syncthreads();
    if (tid < 64) {
      int c0 = jt * 16;
      for (int j = 0; j < 16; ++j) {
        int pp = (c0 + j) / KNN;
        xm[tid][pp] = fmaxf(xm[tid][pp], act2[tid][j]);
      }
    }
    __syncthreads();
  }
  for (int i = tid; i < 64 * 16; i += 128) {
    int row = i >> 4;
    int p = i & 15;
    xout[((size_t)b * 64 + row) * NPT + n0 + p] = xm[row][p];
  }
}

// ---------------------------------------------------------------- stage3: graph_feature(x2) -> conv(w5 64x128) -> max_k
__global__ void stage3_kernel(const float* __restrict__ xin, const int* __restrict__ idx,
                              const float* __restrict__ w5, const float* __restrict__ s5,
                              const float* __restrict__ b5, float* __restrict__ xout) {
  int b = blockIdx.y;
  int n0 = blockIdx.x * 16;
  int tid = threadIdx.x;
  int wave = tid >> 5;
  int lane = tid & 31;
  int l15 = lane & 15;
  int kh = (lane < 16) ? 0 : 2;
  int hoff = (lane < 16) ? 0 : 8;
  __shared__ float act2[64][17];
  __shared__ float xm[64][16];
  for (int i = tid; i < 64 * 16; i += 128) ((float*)xm)[i] = -__builtin_inff();
  const float* xb = xin + (size_t)b * 64 * NPT;
  for (int jt = 0; jt < 20; ++jt) {
    int cl = jt * 16 + l15;
    int p = cl / KNN;
    int k = cl % KNN;
    int n = n0 + p;
    int m = idx[((size_t)b * NPT + n) * KNN + k];
    v8f acc;
#pragma unroll
    for (int i = 0; i < 8; ++i) acc[i] = 0.f;
#pragma unroll
    for (int kk = 0; kk < 16; ++kk) {
      v2f a, bf;
      int mA = wave * 16 + l15;
      a[0] = w5[mA * 128 + kk * 4 + kh + 0];
      a[1] = w5[mA * 128 + kk * 4 + kh + 1];
      int c0 = kk * 4 + kh;
      bf[0] = xb[(size_t)(c0 + 0) * NPT + m] - xb[(size_t)(c0 + 0) * NPT + n];
      bf[1] = xb[(size_t)(c0 + 1) * NPT + m] - xb[(size_t)(c0 + 1) * NPT + n];
      acc = wmma4(a, bf, acc);
    }
#pragma unroll
    for (int kk = 16; kk < 32; ++kk) {
      v2f a, bf;
      int mA = wave * 16 + l15;
      a[0] = w5[mA * 128 + kk * 4 + kh + 0];
      a[1] = w5[mA * 128 + kk * 4 + kh + 1];
      int c0 = (kk - 16) * 4 + kh;
      bf[0] = xb[(size_t)(c0 + 0) * NPT + n];
      bf[1] = xb[(size_t)(c0 + 1) * NPT + n];
      acc = wmma4(a, bf, acc);
    }
    __syncthreads();
#pragma unroll
    for (int r = 0; r < 8; ++r) {
      int row = wave * 16 + r + hoff;
      act2[row][l15] = lrelu(acc[r] * s5[row] + b5[row]);
    }
    __syncthreads();
    if (tid < 64) {
      int c0 = jt * 16;
      for (int j = 0; j < 16; ++j) {
        int pp = (c0 + j) / KNN;
        xm[tid][pp] = fmaxf(xm[tid][pp], act2[tid][j]);
      }
    }
    __syncthreads();
  }
  for (int i = tid; i < 64 * 16; i += 128) {
    int row = i >> 4;
    int p = i & 15;
    xout[((size_t)b * 64 + row) * NPT + n0 + p] = xm[row][p];
  }
}

// ---------------------------------------------------------------- w6: (1024x192) @ concat(x1,x2,x3) with fused partial N-max
__global__ void w6_kernel(const float* __restrict__ x1, const float* __restrict__ x2,
                          const float* __restrict__ x3, const float* __restrict__ w6,
                          const float* __restrict__ s6, const float* __restrict__ b6,
                          float* __restrict__ pm) {
  int b = blockIdx.z;
  int mt0 = blockIdx.y * 64;
  int n0 = blockIdx.x * 16;
  int tid = threadIdx.x;
  int wave = tid >> 5;
  int lane = tid & 31;
  int l15 = lane & 15;
  int kh = (lane < 16) ? 0 : 2;
  int hoff = (lane < 16) ? 0 : 8;
  int n = n0 + l15;
  const float* p1 = x1 + (size_t)b * 64 * NPT;
  const float* p2 = x2 + (size_t)b * 64 * NPT;
  const float* p3 = x3 + (size_t)b * 64 * NPT;
  v8f acc;
#pragma unroll
  for (int i = 0; i < 8; ++i) acc[i] = 0.f;
#pragma unroll
  for (int kk = 0; kk < 48; ++kk) {
    v2f a, bf;
    int mA = mt0 + wave * 16 + l15;
    a[0] = w6[(size_t)mA * 192 + kk * 4 + kh + 0];
    a[1] = w6[(size_t)mA * 192 + kk * 4 + kh + 1];
#pragma unroll
    for (int r = 0; r < 2; ++r) {
      int c = kk * 4 + kh + r;
      const float* src = (c < 64) ? (p1 + (size_t)c * NPT)
                                  : ((c < 128) ? (p2 + (size_t)(c - 64) * NPT)
                                               : (p3 + (size_t)(c - 128) * NPT));
      bf[r] = src[n];
    }
    acc = wmma4(a, bf, acc);
  }
#pragma unroll
  for (int r = 0; r < 8; ++r) {
    int row = mt0 + wave * 16 + r + hoff;
    float v = lrelu(acc[r] * s6[row] + b6[row]);
#pragma unroll
    for (int s = 1; s < 16; s <<= 1) v = fmaxf(v, __shfl_xor(v, s));
    if (l15 == 0) pm[((size_t)b * 1024 + row) * 128 + blockIdx.x] = v;
  }
}

__global__ void zgmax_kernel(const float* __restrict__ pm, float* __restrict__ zg) {
  int row = blockIdx.x;
  int b = blockIdx.y;
  __shared__ float red[128];
  red[threadIdx.x] = pm[((size_t)b * 1024 + row) * 128 + threadIdx.x];
  __syncthreads();
  for (int s = 64; s > 0; s >>= 1) {
    if (threadIdx.x < s) red[threadIdx.x] = fmaxf(red[threadIdx.x], red[threadIdx.x + s]);
    __syncthreads();
  }
  if (threadIdx.x == 0) zg[(size_t)b * 1024 + row] = red[0];
}

__global__ void l7_kernel(const int* __restrict__ cat, const float* __restrict__ w7,
                          const float* __restrict__ s7, const float* __restrict__ b7,
                          float* __restrict__ lbuf) {
  int b = blockIdx.x;
  int o = threadIdx.x;  // 64
  lbuf[b * 64 + o] = lrelu(s7[o] * w7[o * 16 + cat[b]] + b7[o]);
}

// u[b,o] = w8[o, 0:1024] . zg[b] + w8[o, 1024:1088] . l[b]  (column-constant part of w8 GEMM)
__global__ void u8_kernel(const float* __restrict__ w8, const float* __restrict__ zg,
                          const float* __restrict__ lbuf, float* __restrict__ u) {
  int b = blockIdx.x;
  int o = threadIdx.x;  // 256
  const float* wr = w8 + (size_t)o * 1280;
  float acc = 0.f;
  for (int c = 0; c < 1024; ++c) acc += wr[c] * zg[(size_t)b * 1024 + c];
  for (int c = 0; c < 64; ++c) acc += wr[1024 + c] * lbuf[b * 64 + c];
  u[b * 256 + o] = acc;
}

// ---------------------------------------------------------------- fused w8 -> w9 -> w10 -> w11 chain per 16-column tile
__global__ void final_kernel(const float* __restrict__ x1, const float* __restrict__ x2,
                             const float* __restrict__ x3, const float* __restrict__ u,
                             const float* __restrict__ w8, const float* __restrict__ s8,
                             const float* __restrict__ b8, const float* __restrict__ w9,
                             const float* __restrict__ s9, const float* __restrict__ b9,
                             const float* __restrict__ w10, const float* __restrict__ s10,
                             const float* __restrict__ b10, const float* __restrict__ w11,
                             float* __restrict__ out) {
  int b = blockIdx.y;
  int n0 = blockIdx.x * 16;
  int tid = threadIdx.x;  // 256
  int wave = tid >> 5;    // 0..7
  int lane = tid & 31;
  int l15 = lane & 15;
  int kh = (lane < 16) ? 0 : 2;
  int hoff = (lane < 16) ? 0 : 8;
  int n = n0 + l15;
  __shared__ float h8s[256][17];
  __shared__ float h9s[256][17];
  __shared__ float h10s[128][17];
  const float* p1 = x1 + (size_t)b * 64 * NPT;
  const float* p2 = x2 + (size_t)b * 64 * NPT;
  const float* p3 = x3 + (size_t)b * 64 * NPT;
  // stage w8 (K reduced to 192; broadcast rows folded into u)
#pragma unroll
  for (int half = 0; half < 2; ++half) {
    int mt = wave + half * 8;
    v8f acc;
#pragma unroll
    for (int i = 0; i < 8; ++i) acc[i] = 0.f;
#pragma unroll
    for (int kk = 0; kk < 48; ++kk) {
      v2f a, bf;
      int mA = mt * 16 + l15;
      a[0] = w8[(size_t)mA * 1280 + 1088 + kk * 4 + kh + 0];
      a[1] = w8[(size_t)mA * 1280 + 1088 + kk * 4 + kh + 1];
#pragma unroll
      for (int r = 0; r < 2; ++r) {
        int c = kk * 4 + kh + r;
        const float* src = (c < 64) ? (p1 + (size_t)c * NPT)
                                    : ((c < 128) ? (p2 + (size_t)(c - 64) * NPT)
                                                 : (p3 + (size_t)(c - 128) * NPT));
        bf[r] = src[n];
      }
      acc = wmma4(a, bf, acc);
    }
#pragma unroll
    for (int r = 0; r < 8; ++r) {
      int row = mt * 16 + r + hoff;
      h8s[row][l15] = lrelu(s8[row] * (acc[r] + u[b * 256 + row]) + b8[row]);
    }
  }
  __syncthreads();
  // stage w9 (256x256)
#pragma unroll
  for (int half = 0; half < 2; ++half) {
    int mt = wave + half * 8;
    v8f acc;
#pragma unroll
    for (int i = 0; i < 8; ++i) acc[i] = 0.f;
#pragma unroll
    for (int kk = 0; kk < 64; ++kk) {
      v2f a, bf;
      int mA = mt * 16 + l15;
      a[0] = w9[(size_t)mA * 256 + kk * 4 + kh + 0];
      a[1] = w9[(size_t)mA * 256 + kk * 4 + kh + 1];
      bf[0] = h8s[kk * 4 + kh + 0][l15];
      bf[1] = h8s[kk * 4 + kh + 1][l15];
      acc = wmma4(a, bf, acc);
    }
#pragma unroll
    for (int r = 0; r < 8; ++r) {
      int row = mt * 16 + r + hoff;
      h9s[row][l15] = lrelu(s9[row] * acc[r] + b9[row]);
    }
  }
  __syncthreads();
  // stage w10 (128x256)
  {
    int mt = wave;
    v8f acc;
#pragma unroll
    for (int i = 0; i < 8; ++i) acc[i] = 0.f;
#pragma unroll
    for (int kk = 0; kk < 64; ++kk) {
      v2f a, bf;
      int mA = mt * 16 + l15;
      a[0] = w10[(size_t)mA * 256 + kk * 4 + kh + 0];
      a[1] = w10[(size_t)mA * 256 + kk * 4 + kh + 1];
      bf[0] = h9s[kk * 4 + kh + 0][l15];
      bf[1] = h9s[kk * 4 + kh + 1][l15];
      acc = wmma4(a, bf, acc);
    }
#pragma unroll
    for (int r = 0; r < 8; ++r) {
      int row = mt * 16 + r + hoff;
      h10s[row][l15] = lrelu(s10[row] * acc[r] + b10[row]);
    }
  }
  __syncthreads();
  // stage w11 (50x128), waves 0..3, padded to 64 rows
  if (wave < 4) {
    int mt = wave;
    v8f acc;
#pragma unroll
    for (int i = 0; i < 8; ++i) acc[i] = 0.f;
#pragma unroll
    for (int kk = 0; kk < 32; ++kk) {
      v2f a, bf;
      int mA = mt * 16 + l15;
      a[0] = (mA < 50) ? w11[mA * 128 + kk * 4 + kh + 0] : 0.f;
      a[1] = (mA < 50) ? w11[mA * 128 + kk * 4 + kh + 1] : 0.f;
      bf[0] = h10s[kk * 4 + kh + 0][l15];
      bf[1] = h10s[kk * 4 + kh + 1][l15];
      acc = wmma4(a, bf, acc);
    }
#pragma unroll
    for (int r = 0; r < 8; ++r) {
      int o = mt * 16 + r + hoff;
      if (o < 50) out[((size_t)b * NPT + n) * 50 + o] = acc[r];
    }
  }
}

// ================================================================ host launcher
extern "C" void kernel_launch(void* const* d_in, const int* in_sizes, int n_in,
                              void* d_out, int out_size, void* d_ws, size_t ws_size,
                              hipStream_t stream) {
  const float* x = (const float*)d_in[0];
  const int* cat = (const int*)d_in[1];
  const float* R = (const float*)d_in[2];
  const float* pc = (const float*)d_in[3];
  const float* Rc = (const float*)d_in[4];
  const float* w[11];
  for (int i = 0; i < 11; ++i) w[i] = (const float*)d_in[5 + i];
  const float *s[10], *bb[10];
  for (int i = 0; i < 10; ++i) {
    s[i] = (const float*)d_in[16 + 2 * i];
    bb[i] = (const float*)d_in[17 + 2 * i];
  }
  float* out = (float*)d_out;

  char* ws = (char*)d_ws;
  size_t off = 0;
  auto alloc = [&](size_t bytes) -> void* {
    void* p = ws + off;
    off += (bytes + 255) & ~(size_t)255;
    return p;
  };
  float* xg = (float*)alloc((size_t)BDIM * NPT * 3 * 4);
  int* idx1 = (int*)alloc((size_t)BDIM * NPT * KNN * 4);
  int* idx2 = (int*)alloc((size_t)BDIM * NPT * KNN * 4);
  int* idx3 = (int*)alloc((size_t)BDIM * NPT * KNN * 4);
  float* h12 = (float*)alloc((size_t)BDIM * 12 * NKC * 4);
  float* x1 = (float*)alloc((size_t)BDIM * 64 * NPT * 4);
  float* x2 = (float*)alloc((size_t)BDIM * 64 * NPT * 4);
  float* x3 = (float*)alloc((size_t)BDIM * 64 * NPT * 4);
  float* pm = (float*)alloc((size_t)BDIM * 1024 * 128 * 4);
  float* zg = (float*)alloc((size_t)BDIM * 1024 * 4);
  float* lbuf = (float*)alloc((size_t)BDIM * 64 * 4);
  float* u = (float*)alloc((size_t)BDIM * 256 * 4);
  (void)in_sizes; (void)n_in; (void)out_size; (void)ws_size;

  xg_kernel<<<dim3((BDIM * NPT + 255) / 256), 256, 0, stream>>>(x, pc, Rc, xg);
  knn3_kernel<<<dim3(NPT / 256, BDIM), 256, 0, stream>>>(x, idx1);
  build_h12_kernel<<<dim3((BDIM * NPT * KNN + 255) / 256), 256, 0, stream>>>(x, xg, R, idx1, h12);
  stage1_kernel<<<dim3(NPT / 16, BDIM), 128, 0, stream>>>(h12, w[0], s[0], bb[0], w[1], s[1], bb[1], x1);
  knn64_kernel<<<dim3(NPT / 128, BDIM), 128, 0, stream>>>(x1, idx2);
  stage2_kernel<<<dim3(NPT / 16, BDIM), 128, 0, stream>>>(x1, idx2, w[2], s[2], bb[2], w[3], s[3], bb[3], x2);
  knn64_kernel<<<dim3(NPT / 128, BDIM), 128, 0, stream>>>(x2, idx3);
  stage3_kernel<<<dim3(NPT / 16, BDIM), 128, 0, stream>>>(x2, idx3, w[4], s[4], bb[4], x3);
  w6_kernel<<<dim3(NPT / 16, 1024 / 64, BDIM), 128, 0, stream>>>(x1, x2, x3, w[5], s[5], bb[5], pm);
  zgmax_kernel<<<dim3(1024, BDIM), 128, 0, stream>>>(pm, zg);
  l7_kernel<<<dim3(BDIM), 64, 0, stream>>>(cat, w[6], s[6], bb[6], lbuf);
  u8_kernel<<<dim3(BDIM), 256, 0, stream>>>(w[7], zg, lbuf, u);
  final_kernel<<<dim3(NPT / 16, BDIM), 256, 0, stream>>>(x1, x2, x3, u, w[7], s[7], bb[7],
                                                         w[8], s[8], bb[8], w[9], s[9], bb[9],
                                                         w[10], out);
}